// EntropyDecoder_84928683311460
// MI455X (gfx1250) — compile-verified
//
#include <hip/hip_runtime.h>
#include <stdint.h>

// EntropyDecoder for MI455X (gfx1250).
// Inherently serial adaptive arithmetic decode -> single-wave latency-optimized
// loop; CDNA5-specific paths: Tensor Data Mover staging of the packed
// bitstream into LDS + s_wait_tensorcnt, wave32 ballot packing.

#define NSTATES        256
#define MAXWORDS       16384                     // 524288 bits / 32
#define LDS_BITS_BYTES (MAXWORDS * 4)            // 64 KB bitstream
#define LDS_TOTAL      (LDS_BITS_BYTES + 2 * NSTATES * 4)  // + 2 KB counts

typedef uint32_t v4u __attribute__((ext_vector_type(4)));
typedef int      v8i __attribute__((ext_vector_type(8)));
typedef int      v4i __attribute__((ext_vector_type(4)));

// ---------------------------------------------------------------------------
// Pass 1: pack int64-per-bit payload into MSB-first 32-bit words.
// One wave packs one word: lane j loads payload[32w+j] (coalesced 256B/wave),
// ballot collects the 32 predicate bits, __brev flips lane order (LSB-first)
// to MSB-first. Zero-fill past end (reference read_bit returns 0 there).
// code0 == word[0] by construction.
// ---------------------------------------------------------------------------
__global__ void pack_bits_kernel(const long long* __restrict__ bits,
                                 uint32_t* __restrict__ words,
                                 int nb, int nwords) {
  int t    = blockIdx.x * blockDim.x + threadIdx.x;
  int w    = t >> 5;                 // word index = wave index
  int lane = t & 31;
  if (w >= nwords) return;
  long long idx = (long long)w * 32 + lane;
  uint32_t b = (idx < (long long)nb) ? ((uint32_t)bits[idx] & 1u) : 0u;
#if __has_builtin(__builtin_amdgcn_ballot_w32)
  uint32_t mask = __builtin_amdgcn_ballot_w32(b != 0u);
#else
  uint32_t mask = (uint32_t)__ballot(b != 0u);
#endif
  if (lane == 0) words[w] = __brev(mask);   // bit j of payload -> bit (31-j)
}

// ---------------------------------------------------------------------------
// Pass 2: single-wave serial decoder. Bitstream + adaptive counts in LDS.
// Bit decision is division-free: scaled >= p0_freq  <=>  X >= p0_freq*rng.
// FP64 divide kept only for p0 to reproduce the reference's rint() rounding.
// ---------------------------------------------------------------------------
__global__ __launch_bounds__(32) void decode_kernel(
    const uint32_t* __restrict__ words, float* __restrict__ out,
    int nwords, int num_symbols) {
  extern __shared__ uint8_t smem[];
  uint32_t* s_bits = (uint32_t*)smem;                       // 16384 words
  uint32_t* s_c0   = (uint32_t*)(smem + LDS_BITS_BYTES);    // 256 counts
  uint32_t* s_c1   = s_c0 + NSTATES;                        // 256 counts
  int nw = nwords > MAXWORDS ? MAXWORDS : nwords;

#if __has_builtin(__builtin_amdgcn_tensor_load_to_lds)
  {
    // Tensor DMA descriptor (D#), 2-D tile: nw dwords x 1 row, global -> LDS.
    uint64_t ga       = (uint64_t)(uintptr_t)words;
    uint32_t lds_addr = (uint32_t)(uintptr_t)s_bits;        // low 32b = LDS byte offset
    v4u g0;
    g0[0] = 1u;                                             // count=1 (valid), no gather
    g0[1] = lds_addr;                                       // lds_addr [63:32]
    g0[2] = (uint32_t)ga;                                   // global_addr [95:64]
    g0[3] = ((uint32_t)(ga >> 32) & 0x01FFFFFFu)            // global_addr [120:96]
            | 0x80000000u;                                  // type=2 ("image") [127:126]
    uint32_t unw = (uint32_t)nw;
    v8i g1;
    g1[0] = (int)(2u << 16);                  // wg_mask=0, data_size=2 (4 bytes)
    g1[1] = (int)((unw & 0xFFFFu) << 16);     // tensor_dim0 [15:0] at bits 63:48
    g1[2] = (int)(((unw >> 16) & 0xFFFFu)     // tensor_dim0 [31:16]
                  | (1u << 16));              // tensor_dim1 = 1 (low half)
    g1[3] = (int)((unw & 0xFFFFu) << 16);     // tensor_dim1 hi=0, tile_dim0 = nw
    g1[4] = 1;                                // tile_dim1 = 1, tile_dim2 = 0
    g1[5] = (int)unw;                         // tensor_dim0_stride low 32
    g1[6] = 0;                                // stride hi / dim1_stride low
    g1[7] = 0;
    v4i gz4 = {0, 0, 0, 0};                   // groups 2/3 unused (2-D tensor)
    v8i gz8 = {0, 0, 0, 0, 0, 0, 0, 0};       // extra group (clang-23 6-arg form)
    __builtin_amdgcn_tensor_load_to_lds(g0, g1, gz4, gz4, gz8, 0);
  }
  // Initialize the Laplace counts while the TDM streams the bitstream.
  for (int i = threadIdx.x; i < NSTATES; i += 32) { s_c0[i] = 1u; s_c1[i] = 1u; }
#if __has_builtin(__builtin_amdgcn_s_wait_tensorcnt)
  __builtin_amdgcn_s_wait_tensorcnt(0);
#else
  asm volatile("s_wait_tensorcnt 0" ::: "memory");
#endif
#else
  // Fallback: cooperative wave load of the bitstream.
  for (int i = threadIdx.x; i < nw; i += 32) s_bits[i] = words[i];
  for (int i = threadIdx.x; i < NSTATES; i += 32) { s_c0[i] = 1u; s_c1[i] = 1u; }
#endif
  __syncthreads();

  if (threadIdx.x != 0) return;

  uint32_t low = 0u, high = 0xFFFFFFFFu;
  uint32_t code = (nw > 0) ? s_bits[0] : 0u;   // first 32 payload bits, MSB first
  int      widx = 1;
  uint32_t cur  = (nw > 1) ? s_bits[1] : 0u;   // 32-bit register bit-cache
  int      curbits = 32;
  uint32_t state = 0u;

  for (int s = 0; s < num_symbols; ++s) {
    uint32_t n0 = s_c0[state];
    uint32_t n1 = s_c1[state];
    uint32_t total = n0 + n1;

    // p0_freq = clip(rint((n0/total)*4094) + 1, 1, 4095)  — matches reference
    // double-rounding exactly (f64 divide, then f64 multiply, then RNE rint).
    double p0 = (double)n0 / (double)total;
    int pf = (int)rint(p0 * 4094.0) + 1;
    pf = pf < 1 ? 1 : (pf > 4095 ? 4095 : pf);

    uint64_t rng = (uint64_t)(high - low) + 1ull;                  // <= 2^32
    uint64_t X   = (((uint64_t)(code - low) + 1ull) << 12) - 1ull; // (code-low+1)*4096-1
    uint64_t Y   = (uint64_t)(uint32_t)pf * rng;                   // p0_freq*rng
    uint32_t bit = (X >= Y) ? 1u : 0u;            // scaled >= p0_freq, no division
    uint32_t split = low + (uint32_t)(Y >> 12);   // low + rng*p0_freq/4096

    if (bit) low = split; else high = split - 1u;

    // E1/E2/E3 renormalization.
    for (;;) {
      uint32_t sub;
      if (high < 0x80000000u)                               sub = 0u;
      else if (low >= 0x80000000u)                          sub = 0x80000000u;
      else if (low >= 0x40000000u && high < 0xC0000000u)    sub = 0x40000000u;
      else break;
      low  = (low  - sub) << 1;
      high = ((high - sub) << 1) | 1u;
      if (curbits == 0) {
        ++widx;
        cur = (widx < nw) ? s_bits[widx] : 0u;   // 0 past end of payload
        curbits = 32;
      }
      code = ((code - sub) << 1) | (cur >> 31);
      cur <<= 1;
      --curbits;
    }

    s_c0[state] = n0 + (1u - bit);
    s_c1[state] = n1 + bit;
    state = ((state << 1) | bit) & (NSTATES - 1);
    out[s] = bit ? 1.0f : -1.0f;                 // {0,1} -> {-1,+1}
  }
}

// ---------------------------------------------------------------------------
extern "C" void kernel_launch(void* const* d_in, const int* in_sizes, int n_in,
                              void* d_out, int out_size, void* d_ws, size_t ws_size,
                              hipStream_t stream) {
  const long long* payload = (const long long*)d_in[0];  // int64 payload_bits
  int nb = in_sizes[0];                                   // 524288 bits
  int num_symbols = out_size;                             // 262144 symbols
  int nwords = (nb + 31) / 32;
  if (nwords > MAXWORDS) nwords = MAXWORDS;

  // Stage packed words in d_ws; if it's too small, borrow d_out (safe: the
  // decoder copies all words into LDS before writing any output).
  uint32_t* staging = (ws_size >= (size_t)nwords * 4u)
                          ? (uint32_t*)d_ws : (uint32_t*)d_out;

  const int threads = 256;
  long long total_threads = (long long)nwords * 32;       // one wave per word
  int blocks = (int)((total_threads + threads - 1) / threads);
  pack_bits_kernel<<<blocks, threads, 0, stream>>>(payload, staging, nb, nwords);
  decode_kernel<<<1, 32, LDS_TOTAL, stream>>>(staging, (float*)d_out,
                                              nwords, num_symbols);
}